// MultiHeadAttention_60584808677786
// MI455X (gfx1250) — compile-verified
//
#include <hip/hip_runtime.h>
#include <hip/hip_bf16.h>

// Graph transformer layer for MI455X (gfx1250), fp32 end-to-end.
// GEMMs: V_WMMA_F32_16X16X4_F32 fed from LDS, staged with
// GLOBAL_LOAD_ASYNC_TO_LDS_B128 (ASYNCcnt) + s_wait_asynccnt.

typedef float v2f __attribute__((ext_vector_type(2)));
typedef float v8f __attribute__((ext_vector_type(8)));

#define NN   10000   // nodes
#define EE   80000   // edges
#define DD   128     // d_model
#define HH   8       // heads
#define HD   1024    // H*DH
#define DFFN 512

// ---------------------------------------------------------------------------
// CDNA5 async copy: 16B global -> LDS, tracked by ASYNCcnt.
// LDS offset = low 32 bits of the generic shared-pointer (AS3 offset).
// ---------------------------------------------------------------------------
__device__ __forceinline__ void async_copy_b128(const float* gptr, const float* sptr) {
    asm volatile("global_load_async_to_lds_b128 %0, %1, off"
                 :
                 : "v"((unsigned int)(unsigned long long)(uintptr_t)sptr),
                   "v"((unsigned long long)(uintptr_t)gptr)
                 : "memory");
}
__device__ __forceinline__ void wait_async_zero() {
    asm volatile("s_wait_asynccnt 0x0" ::: "memory");
}

__device__ __forceinline__ void zero_acc(v8f acc[4]) {
#pragma unroll
    for (int t = 0; t < 4; ++t)
        acc[t] = (v8f){0.f,0.f,0.f,0.f,0.f,0.f,0.f,0.f};
}

// ---------------------------------------------------------------------------
// Block-cooperative tile GEMM: C(16 x COLS) += A(16 x K) * B(K x COLS)
// Block = COLS/64 waves; wave w owns cols [64w, 64w+64).
// K consumed in 32-wide slices staged in LDS via async b128 copies.
// WMMA f32 layout: A lane L -> M=L&15, K={2*(L>>4),2*(L>>4)+1};
//                  B lane L -> N=L&15, same K pairing;
//                  D VGPR v -> M=v+8*(L>>4), N=L&15.
// Padding: Asm stride 36 (banks 4m, conflict-free b64);
//          Bsm stride COLS+16 (half-waves 32 banks apart, conflict-free b32).
// ---------------------------------------------------------------------------
template<int COLS>
__device__ __forceinline__ void gemm_block_tile(
    const float* __restrict__ A, const float* __restrict__ B,
    int lda, int ldb, int Kdim, int row0, int colB0, v8f acc[4])
{
    constexpr int WAVES = COLS / 64;
    constexpr int NTHR  = 32 * WAVES;
    constexpr int BSTR  = COLS + 16;
    __shared__ __attribute__((aligned(16))) float Asm[16 * 36];
    __shared__ __attribute__((aligned(16))) float Bsm[32 * BSTR];

    const int tid  = threadIdx.x;
    const int lane = tid & 31;
    const int wave = tid >> 5;
    const int m    = lane & 15;
    const int kp   = lane >> 4;          // 0 or 1

    for (int kk = 0; kk < Kdim; kk += 32) {
        // Stage A slice: 16 rows x 32 floats (128 float4)
#pragma unroll
        for (int i = tid; i < 16 * 8; i += NTHR) {
            const int r = i >> 3, q = i & 7;
            async_copy_b128(A + (size_t)(row0 + r) * lda + kk + q * 4,
                            &Asm[r * 36 + q * 4]);
        }
        // Stage B slice: 32 rows x COLS floats
#pragma unroll
        for (int i = tid; i < 32 * (COLS / 4); i += NTHR) {
            const int r = i / (COLS / 4), q = i % (COLS / 4);
            async_copy_b128(B + (size_t)(kk + r) * ldb + colB0 + q * 4,
                            &Bsm[r * BSTR + q * 4]);
        }
        wait_async_zero();       // this wave's copies landed in LDS
        __syncthreads();         // all waves' copies visible

#pragma unroll
        for (int k2 = 0; k2 < 32; k2 += 4) {
            v2f a = *(const v2f*)&Asm[m * 36 + k2 + 2 * kp];
#pragma unroll
            for (int t = 0; t < 4; ++t) {
                const int cw = wave * 64 + 16 * t + m;
                v2f b;
                b.x = Bsm[(k2 + 2 * kp) * BSTR + cw];
                b.y = Bsm[(k2 + 2 * kp + 1) * BSTR + cw];
                acc[t] = __builtin_amdgcn_wmma_f32_16x16x4_f32(
                    false, a, false, b, (short)0, acc[t], false, false);
            }
        }
        __syncthreads();         // protect LDS before next slice overwrite
    }
}

// ---------------------------------------------------------------------------
// QKV projection: Out(N x 1024) = feat(N x 128) @ W(128 x 1024)
// grid (625, 4), block 128
// ---------------------------------------------------------------------------
__global__ void k_proj1024(const float* __restrict__ A, const float* __restrict__ W,
                           float* __restrict__ Out)
{
    const int lane = threadIdx.x & 31;
    const int wave = threadIdx.x >> 5;
    const int row0 = blockIdx.x * 16;
    const int colB0 = blockIdx.y * 256;
    v8f acc[4]; zero_acc(acc);
    gemm_block_tile<256>(A, W, DD, HD, DD, row0, colB0, acc);
    const int n = lane & 15, kp = lane >> 4;
    const int col0 = colB0 + wave * 64;
#pragma unroll
    for (int t = 0; t < 4; ++t)
#pragma unroll
        for (int v = 0; v < 8; ++v)
            Out[(size_t)(row0 + v + 8 * kp) * HD + col0 + 16 * t + n] = acc[t][v];
}

// ---------------------------------------------------------------------------
// Per-edge logits -> exp -> segment-sum denominator. Clamp(+-5) bounds
// exp(u) in [6.7e-3, 148], so the max-shift is unnecessary numerically.
// One wave per edge; lane covers 32 contiguous floats (head = lane>>2).
// ---------------------------------------------------------------------------
__global__ void k_edge_softmax_num(const float* __restrict__ Q, const float* __restrict__ Kn,
                                   const int* __restrict__ src, const int* __restrict__ dst,
                                   float* __restrict__ ex, float* __restrict__ den)
{
    const int e = blockIdx.x * (blockDim.x >> 5) + (threadIdx.x >> 5);
    if (e >= EE) return;
    const int lane = threadIdx.x & 31;
    const int s = src[e], d = dst[e];
    const float4* q4 = (const float4*)(Q  + (size_t)d * HD) + lane * 8;
    const float4* k4 = (const float4*)(Kn + (size_t)s * HD) + lane * 8;
    float p = 0.f;
#pragma unroll
    for (int j = 0; j < 8; ++j) {
        float4 q = q4[j], k = k4[j];
        p += q.x * k.x + q.y * k.y + q.z * k.z + q.w * k.w;
    }
    p += __shfl_xor(p, 1, 32);   // reduce the 4 lanes of one head
    p += __shfl_xor(p, 2, 32);
    if ((lane & 3) == 0) {
        const int h = lane >> 2;
        float u = p * 0.08838834764831845f;        // 1/sqrt(128)
        u = fminf(fmaxf(u, -5.f), 5.f);
        float v = expf(u);
        ex[(size_t)e * HH + h] = v;
        atomicAdd(den + (size_t)d * HH + h, v);    // global_atomic_add_f32
    }
}

// ---------------------------------------------------------------------------
// agg[dst] += (ex/den) * V[src] ; one wave per edge, lane -> 32 floats
// ---------------------------------------------------------------------------
__global__ void k_aggregate(const float* __restrict__ V,
                            const int* __restrict__ src, const int* __restrict__ dst,
                            const float* __restrict__ ex, const float* __restrict__ den,
                            float* __restrict__ agg)
{
    const int e = blockIdx.x * (blockDim.x >> 5) + (threadIdx.x >> 5);
    if (e >= EE) return;
    const int lane = threadIdx.x & 31;
    const int s = src[e], d = dst[e];
    const int h = lane >> 2;
    const float a = ex[(size_t)e * HH + h] / den[(size_t)d * HH + h];
    const float4* v4 = (const float4*)(V + (size_t)s * HD) + lane * 8;
    float* o = agg + (size_t)d * HD + lane * 32;
#pragma unroll
    for (int j = 0; j < 8; ++j) {
        float4 x = v4[j];
        atomicAdd(o + 4 * j + 0, a * x.x);
        atomicAdd(o + 4 * j + 1, a * x.y);
        atomicAdd(o + 4 * j + 2, a * x.z);
        atomicAdd(o + 4 * j + 3, a * x.w);
    }
}

// ---------------------------------------------------------------------------
// tmp1 = feat + agg(N x 1024) @ Wo(1024 x 128) ; grid 625, block 64 (2 waves)
// ---------------------------------------------------------------------------
__global__ void k_oproj(const float* __restrict__ agg, const float* __restrict__ Wo,
                        const float* __restrict__ feat, float* __restrict__ tmp1)
{
    const int lane = threadIdx.x & 31;
    const int wave = threadIdx.x >> 5;
    const int row0 = blockIdx.x * 16;
    v8f acc[4]; zero_acc(acc);
    gemm_block_tile<128>(agg, Wo, HD, DD, HD, row0, 0, acc);
    const int n = lane & 15, kp = lane >> 4;
    const int col0 = wave * 64;
#pragma unroll
    for (int t = 0; t < 4; ++t)
#pragma unroll
        for (int v = 0; v < 8; ++v) {
            const int r = row0 + v + 8 * kp, c = col0 + 16 * t + n;
            tmp1[(size_t)r * DD + c] = acc[t][v] + feat[(size_t)r * DD + c];
        }
}

// ---------------------------------------------------------------------------
// z = relu(uh(N x 128) @ W1(128 x 512) + bf1) ; grid (625,2), block 128
// ---------------------------------------------------------------------------
__global__ void k_ffn1(const float* __restrict__ uh, const float* __restrict__ W1,
                       const float* __restrict__ bf1, float* __restrict__ z)
{
    const int lane = threadIdx.x & 31;
    const int wave = threadIdx.x >> 5;
    const int row0 = blockIdx.x * 16;
    const int colB0 = blockIdx.y * 256;
    v8f acc[4]; zero_acc(acc);
    gemm_block_tile<256>(uh, W1, DD, DFFN, DD, row0, colB0, acc);
    const int n = lane & 15, kp = lane >> 4;
    const int col0 = colB0 + wave * 64;
#pragma unroll
    for (int t = 0; t < 4; ++t)
#pragma unroll
        for (int v = 0; v < 8; ++v) {
            const int r = row0 + v + 8 * kp, c = col0 + 16 * t + n;
            z[(size_t)r * DFFN + c] = fmaxf(acc[t][v] + bf1[c], 0.f);
        }
}

// ---------------------------------------------------------------------------
// tmp2 = uh + z(N x 512) @ W2(512 x 128) + bf2 ; grid 625, block 64
// ---------------------------------------------------------------------------
__global__ void k_ffn2(const float* __restrict__ z, const float* __restrict__ W2,
                       const float* __restrict__ bf2, const float* __restrict__ uh,
                       float* __restrict__ tmp2)
{
    const int lane = threadIdx.x & 31;
    const int wave = threadIdx.x >> 5;
    const int row0 = blockIdx.x * 16;
    v8f acc[4]; zero_acc(acc);
    gemm_block_tile<128>(z, W2, DFFN, DD, DFFN, row0, 0, acc);
    const int n = lane & 15, kp = lane >> 4;
    const int col0 = wave * 64;
#pragma unroll
    for (int t = 0; t < 4; ++t)
#pragma unroll
        for (int v = 0; v < 8; ++v) {
            const int r = row0 + v + 8 * kp, c = col0 + 16 * t + n;
            tmp2[(size_t)r * DD + c] = acc[t][v] + bf2[c] + uh[(size_t)r * DD + c];
        }
}

// ---------------------------------------------------------------------------
// Row-wise LayerNorm over 128 cols; one wave per row (lane -> float4)
// ---------------------------------------------------------------------------
__global__ void k_layernorm(const float* __restrict__ X, const float* __restrict__ g,
                            const float* __restrict__ b, float* __restrict__ Y)
{
    const int row = blockIdx.x * (blockDim.x >> 5) + (threadIdx.x >> 5);
    if (row >= NN) return;
    const int lane = threadIdx.x & 31;
    float4 x = ((const float4*)(X + (size_t)row * DD))[lane];
    float s = x.x + x.y + x.z + x.w;
#pragma unroll
    for (int o = 16; o; o >>= 1) s += __shfl_xor(s, o, 32);
    const float mu = s * (1.f / 128.f);
    const float d0 = x.x - mu, d1 = x.y - mu, d2 = x.z - mu, d3 = x.w - mu;
    float var = d0 * d0 + d1 * d1 + d2 * d2 + d3 * d3;
#pragma unroll
    for (int o = 16; o; o >>= 1) var += __shfl_xor(var, o, 32);
    const float r = rsqrtf(var * (1.f / 128.f) + 1e-5f);
    float4 gg = ((const float4*)g)[lane];
    float4 bb = ((const float4*)b)[lane];
    float4 y;
    y.x = d0 * r * gg.x + bb.x;
    y.y = d1 * r * gg.y + bb.y;
    y.z = d2 * r * gg.z + bb.z;
    y.w = d3 * r * gg.w + bb.w;
    ((float4*)(Y + (size_t)row * DD))[lane] = y;
}

__global__ void k_zero(float* __restrict__ p, int n) {
    const int i = blockIdx.x * blockDim.x + threadIdx.x;
    if (i < n) p[i] = 0.f;
}

// ---------------------------------------------------------------------------
extern "C" void kernel_launch(void* const* d_in, const int* in_sizes, int n_in,
                              void* d_out, int out_size, void* d_ws, size_t ws_size,
                              hipStream_t stream)
{
    const float* feat = (const float*)d_in[0];
    const int*   src  = (const int*)d_in[1];
    const int*   dst  = (const int*)d_in[2];
    const float* Wq   = (const float*)d_in[3];
    const float* Wk   = (const float*)d_in[4];
    const float* Wv   = (const float*)d_in[5];
    const float* Wo   = (const float*)d_in[6];
    const float* g1   = (const float*)d_in[7];
    const float* b1   = (const float*)d_in[8];
    const float* W1   = (const float*)d_in[9];
    const float* bf1  = (const float*)d_in[10];
    const float* W2   = (const float*)d_in[11];
    const float* bf2  = (const float*)d_in[12];
    const float* g2   = (const float*)d_in[13];
    const float* b2   = (const float*)d_in[14];
    float* out = (float*)d_out;

    // Workspace layout (~141.5 MB), with buffer reuse:
    float* Q    = (float*)d_ws;                 // N*1024
    float* Kn   = Q    + (size_t)NN * HD;       // N*1024
    float* Vn   = Kn   + (size_t)NN * HD;       // N*1024
    float* ex   = Vn   + (size_t)NN * HD;       // E*8
    float* den  = ex   + (size_t)EE * HH;       // N*8
    float* tmp1 = den  + (size_t)NN * HH;       // N*128
    float* uh   = tmp1 + (size_t)NN * DD;       // N*128
    float* tmp2 = uh   + (size_t)NN * DD;       // N*128
    float* agg  = Q;    // Q dead after edge-logits kernel
    float* z    = Kn;   // K dead after edge-logits kernel (512 <= 1024 cols)

    const dim3 gProj(NN / 16, HD / 256);        // (625, 4)
    k_proj1024<<<gProj, 128, 0, stream>>>(feat, Wq, Q);
    k_proj1024<<<gProj, 128, 0, stream>>>(feat, Wk, Kn);
    k_proj1024<<<gProj, 128, 0, stream>>>(feat, Wv, Vn);

    k_zero<<<(NN * HH + 255) / 256, 256, 0, stream>>>(den, NN * HH);
    k_edge_softmax_num<<<EE / 8, 256, 0, stream>>>(Q, Kn, src, dst, ex, den);

    k_zero<<<(NN * HD + 255) / 256, 256, 0, stream>>>(agg, NN * HD);
    k_aggregate<<<EE / 8, 256, 0, stream>>>(Vn, src, dst, ex, den, agg);

    k_oproj<<<NN / 16, 64, 0, stream>>>(agg, Wo, feat, tmp1);
    k_layernorm<<<NN / 8, 256, 0, stream>>>(tmp1, g1, b1, uh);

    k_ffn1<<<dim3(NN / 16, DFFN / 256), 128, 0, stream>>>(uh, W1, bf1, z);
    k_ffn2<<<NN / 16, 64, 0, stream>>>(z, W2, bf2, uh, tmp2);
    k_layernorm<<<NN / 8, 256, 0, stream>>>(tmp2, g2, b2, out);
}